// ChamferReward_14757507629948
// MI455X (gfx1250) — compile-verified
//
#include <hip/hip_runtime.h>
#include <math.h>

typedef __attribute__((ext_vector_type(2))) float v2f;
typedef __attribute__((ext_vector_type(8))) float v8f;

#define NP 1024
#define FD 10

__device__ __forceinline__ unsigned long long umin64(unsigned long long a,
                                                     unsigned long long b) {
  return a < b ? a : b;
}

__global__ __launch_bounds__(128) void chamfer_zero(float* out, int n) {
  int i = threadIdx.x + blockIdx.x * blockDim.x;
  if (i < n) out[i] = 0.0f;
}

// One workgroup per (b, v). 256 threads = 8 waves (wave32).
__global__ __launch_bounds__(256) void chamfer_kernel(
    const float* __restrict__ ag,    // achieved_goal (state) [128,4,1024,10]
    const float* __restrict__ dg,    // desired_goal (goal)
    const float* __restrict__ nmean, // [10]
    const float* __restrict__ nstd,  // [10]
    float* __restrict__ out) {       // [128]
  __shared__ float s_vis[NP * 4];
  __shared__ float g_vis[NP * 4];
  __shared__ float s_xx[NP];
  __shared__ float g_xx[NP];
  __shared__ unsigned long long colmin[NP]; // min over state n, per goal m (min_d1/idx1)
  __shared__ unsigned long long rowmin[NP]; // min over goal m, per state n (min_d2/idx2)
  __shared__ float red[256];

  const int tid = threadIdx.x;
  const int bv = blockIdx.x; // b*4 + v
  const int b = bv >> 2;

  const float me0 = nmean[0], me1 = nmean[1];
  const float st0 = nstd[0], st1 = nstd[1];
  const float me5 = nmean[5], me6 = nmean[6], me7 = nmean[7], me8 = nmean[8];
  const float st5 = nstd[5], st6 = nstd[6], st7 = nstd[7], st8 = nstd[8];

  const float* __restrict__ agp = ag + (size_t)bv * (NP * FD);
  const float* __restrict__ dgp = dg + (size_t)bv * (NP * FD);

  // Phase 0: normalize features 5..8 into LDS, precompute squared norms,
  // init packed min arrays.
  for (int p = tid; p < NP; p += 256) {
    const float* a = agp + p * FD;
    float a5 = a[5] * st5 + me5, a6 = a[6] * st6 + me6;
    float a7 = a[7] * st7 + me7, a8 = a[8] * st8 + me8;
    s_vis[p * 4 + 0] = a5; s_vis[p * 4 + 1] = a6;
    s_vis[p * 4 + 2] = a7; s_vis[p * 4 + 3] = a8;
    s_xx[p] = a5 * a5 + a6 * a6 + a7 * a7 + a8 * a8;

    const float* g = dgp + p * FD;
    float g5 = g[5] * st5 + me5, g6 = g[6] * st6 + me6;
    float g7 = g[7] * st7 + me7, g8 = g[8] * st8 + me8;
    g_vis[p * 4 + 0] = g5; g_vis[p * 4 + 1] = g6;
    g_vis[p * 4 + 2] = g7; g_vis[p * 4 + 3] = g8;
    g_xx[p] = g5 * g5 + g6 * g6 + g7 * g7 + g8 * g8;

    colmin[p] = ~0ull;
    rowmin[p] = ~0ull;
  }
  __syncthreads();

  // Phase 1: 64x64 tiles of 16x16 via V_WMMA_F32_16X16X4_F32 (K = 4 features).
  const int lane = tid & 31;
  const int wid = tid >> 5;       // wave id 0..7
  const int lane16 = lane & 15;
  const int hi = lane >> 4;       // lane group selector
  const int kbase = hi << 1;      // K base: 0 or 2 (ISA 16x4 f32 A layout)

  for (int rt = wid; rt < 64; rt += 8) { // row-tile strip per wave (uniform)
    const int arow = rt * 16 + lane16;
    v2f a;
    a.x = s_vis[arow * 4 + kbase];
    a.y = s_vis[arow * 4 + kbase + 1];

    float sxxr[8];
#pragma unroll
    for (int r = 0; r < 8; ++r) sxxr[r] = s_xx[rt * 16 + r + 8 * hi];

    unsigned long long rowm[8];
#pragma unroll
    for (int r = 0; r < 8; ++r) rowm[r] = ~0ull;

    for (int ct = 0; ct < 64; ++ct) {
      const int bcol = ct * 16 + lane16;
      v2f bb; // B 4x16 = A-layout of (goal)^T
      bb.x = g_vis[bcol * 4 + kbase];
      bb.y = g_vis[bcol * 4 + kbase + 1];

      v8f c = {};
      // D = A(16x4) x B(4x16): per-lane reg r -> P row M=r+8*hi, col N=lane16
      v8f d = __builtin_amdgcn_wmma_f32_16x16x4_f32(
          false, a, false, bb, (short)0, c, false, false);

      const float gyy = g_xx[bcol];
      unsigned long long colpart = ~0ull;
#pragma unroll
      for (int r = 0; r < 8; ++r) {
        float p = fmaxf(sxxr[r] + gyy - 2.0f * d[r], 0.0f);
        unsigned long long key = ((unsigned long long)__float_as_uint(p)) << 32;
        colpart = umin64(colpart, key | (unsigned)(rt * 16 + r + 8 * hi));
        rowm[r] = umin64(rowm[r], key | (unsigned)bcol);
      }
      atomicMin(&colmin[bcol], colpart); // ds_min_u64
    }
#pragma unroll
    for (int r = 0; r < 8; ++r)
      atomicMin(&rowmin[rt * 16 + r + 8 * hi], rowm[r]);
  }
  __syncthreads();

  // Phase 2: XY chamfer terms from packed (dist,idx) keys.
  float total = 0.0f;
  for (int m = tid; m < NP; m += 256) {
    unsigned long long k = colmin[m];
    float dmin = __uint_as_float((unsigned)(k >> 32));
    int n = (int)(k & 0x3FFu);
    float gx = dgp[m * FD + 0] * st0 + me0, gy = dgp[m * FD + 1] * st1 + me1;
    float sx = agp[n * FD + 0] * st0 + me0, sy = agp[n * FD + 1] * st1 + me1;
    float dx = gx - sx, dy = gy - sy;
    total += (dmin > 6.0f) ? 1.0f : sqrtf(dx * dx + dy * dy);
  }
  for (int n = tid; n < NP; n += 256) {
    unsigned long long k = rowmin[n];
    float dmin = __uint_as_float((unsigned)(k >> 32));
    int m = (int)(k & 0x3FFu);
    float sx = agp[n * FD + 0] * st0 + me0, sy = agp[n * FD + 1] * st1 + me1;
    float gx = dgp[m * FD + 0] * st0 + me0, gy = dgp[m * FD + 1] * st1 + me1;
    float dx = sx - gx, dy = sy - gy;
    total += (dmin > 6.0f) ? 1.0f : sqrtf(dx * dx + dy * dy);
  }

  red[tid] = total;
  __syncthreads();
  for (int s = 128; s > 0; s >>= 1) {
    if (tid < s) red[tid] += red[tid + s];
    __syncthreads();
  }
  // rv = -(sum_g2s + sum_s2g)/(2*1024); mean over 4 views -> /4  => /8192
  if (tid == 0) atomicAdd(&out[b], -red[0] * (1.0f / 8192.0f));
}

extern "C" void kernel_launch(void* const* d_in, const int* in_sizes, int n_in,
                              void* d_out, int out_size, void* d_ws,
                              size_t ws_size, hipStream_t stream) {
  const float* ag = (const float*)d_in[0];
  const float* dg = (const float*)d_in[1];
  const float* nmean = (const float*)d_in[2];
  const float* nstd = (const float*)d_in[3];
  float* out = (float*)d_out;

  chamfer_zero<<<1, 128, 0, stream>>>(out, out_size);
  chamfer_kernel<<<512, 256, 0, stream>>>(ag, dg, nmean, nstd, out);
}